// PipelinedGemm_45664092291686
// MI455X (gfx1250) — compile-verified
//
#include <hip/hip_runtime.h>
#include <cstdint>

typedef float v2f __attribute__((ext_vector_type(2)));
typedef float v8f __attribute__((ext_vector_type(8)));
typedef unsigned int v4u __attribute__((ext_vector_type(4)));
typedef int v8i __attribute__((ext_vector_type(8)));
typedef int v4i __attribute__((ext_vector_type(4)));

#define M_DIM 8192
#define N_DIM 8192
#define K_DIM 256
#define BM 128
#define BN 128
#define KC 128
#define LDA_S 132   /* 128 + 4 pad floats (TDM pad)  -> A row loads conflict-free */
#define LDB_S 136   /* 128 + 8 pad floats (TDM pad)  -> B half-wave banks disjoint */

// ---- Tensor Data Mover descriptor builders (CDNA5 ISA) ----
static __device__ __forceinline__ v4u tdm_group0(uint32_t lds_addr, const void* gptr) {
  uint64_t ga = (uint64_t)(uintptr_t)gptr;
  v4u g;
  g[0] = 1u;                                    // count=1 (valid), user mode, no gather
  g[1] = lds_addr;                              // LDS byte address
  g[2] = (uint32_t)ga;                          // global_addr[31:0]
  g[3] = (uint32_t)((ga >> 32) & 0x01FFFFFFu)   // global_addr[56:32]
       | (2u << 30);                            // type = 2 ("image")
  return g;
}

static __device__ __forceinline__ v8i tdm_group1(uint32_t tensor_d0, uint32_t tensor_d1,
                                                 uint32_t tile_d0, uint32_t tile_d1,
                                                 uint64_t stride0_elems,
                                                 uint32_t pad_interval_code,
                                                 uint32_t pad_amount_code) {
  uint32_t w0 = (2u << 16)                  // data_size = 2 -> 4 bytes
              | (1u << 20)                  // pad_enable (LDS row padding on load)
              | (pad_interval_code << 22)   // pad every 2^(c+1) DWORDs
              | (pad_amount_code  << 25);   // pad (c+1) DWORDs
  uint32_t w1 = (tensor_d0 & 0xFFFFu) << 16;                         // abar=0 | td0[15:0]
  uint32_t w2 = ((tensor_d0 >> 16) & 0xFFFFu) | ((tensor_d1 & 0xFFFFu) << 16);
  uint32_t w3 = ((tensor_d1 >> 16) & 0xFFFFu) | ((tile_d0 & 0xFFFFu) << 16);
  uint32_t w4 = tile_d1 & 0xFFFFu;                                   // tile_dim2 = 0 (2D)
  uint32_t w5 = (uint32_t)stride0_elems;                             // dim0 stride[31:0]
  uint32_t w6 = (uint32_t)((stride0_elems >> 32) & 0xFFFFu);         // dim0 stride[47:32]
  uint32_t w7 = 0;                                                   // dim1 stride unused (2D)
  v8i g;
  g[0]=(int)w0; g[1]=(int)w1; g[2]=(int)w2; g[3]=(int)w3;
  g[4]=(int)w4; g[5]=(int)w5; g[6]=(int)w6; g[7]=(int)w7;
  return g;
}

// One K-chunk of WMMA work: 32 k-steps of 16x16x4 over a 32x64 per-wave tile.
static __device__ __forceinline__ void compute_chunk(const float* __restrict__ as,
                                                     const float* __restrict__ bs,
                                                     int waveM, int waveN, int l15, int kh,
                                                     v8f acc[2][4]) {
  for (int k0 = 0; k0 < KC; k0 += 4) {
    // A fragment: 16x4 f32, 2 VGPRs/lane: lanes 0-15 -> K={k0,k0+1}, 16-31 -> K={k0+2,k0+3}
    v2f afrag[2];
#pragma unroll
    for (int mi = 0; mi < 2; ++mi) {
      const float* p = &as[(waveM * 32 + mi * 16 + l15) * LDA_S + k0 + 2 * kh];
      afrag[mi].x = p[0];
      afrag[mi].y = p[1];
    }
    // B fragment: 4x16 f32, VGPR v holds rows K=v (lanes 0-15) and K=v+2 (lanes 16-31)
    v2f bfrag[4];
#pragma unroll
    for (int ni = 0; ni < 4; ++ni) {
      const int col = waveN * 64 + ni * 16 + l15;
      bfrag[ni].x = bs[(k0 + 2 * kh + 0) * LDB_S + col];
      bfrag[ni].y = bs[(k0 + 2 * kh + 1) * LDB_S + col];
    }
#pragma unroll
    for (int mi = 0; mi < 2; ++mi)
#pragma unroll
      for (int ni = 0; ni < 4; ++ni)
        acc[mi][ni] = __builtin_amdgcn_wmma_f32_16x16x4_f32(
            false, afrag[mi], false, bfrag[ni],
            (short)0, acc[mi][ni], false, false);
  }
}

__global__ __launch_bounds__(256) void
PipelinedGemm_45664092291686_kernel(const float* __restrict__ A,
                                    const float* __restrict__ B,
                                    const int*   __restrict__ wsz,
                                    float*       __restrict__ C) {
  // Double-buffered tiles: both K-chunks resident -> all TDM issued up front.
  __shared__ __align__(16) float As[2][BM * LDA_S];  // 2 x 128 rows, stride 132 floats
  __shared__ __align__(16) float Bs[2][KC * LDB_S];  // 2 x 128 k-rows, stride 136 floats

  const int tid   = threadIdx.x;
  const int lane  = tid & 31;
  const int wave  = tid >> 5;
  const int waveM = wave >> 1;      // 0..3 -> 32-row band
  const int waveN = wave & 1;       // 0..1 -> 64-col band
  const int l15   = lane & 15;
  const int kh    = lane >> 4;      // half-wave selector (K pair / M+8)

  const int mBase = blockIdx.y * BM;
  const int nBase = blockIdx.x * BN;

  const int   w   = wsz[0];
  const float inc = 0.5f * (float)w * (float)(w + 1);   // sum over ranks

  v8f acc[2][4];
#pragma unroll
  for (int mi = 0; mi < 2; ++mi)
#pragma unroll
    for (int ni = 0; ni < 4; ++ni)
#pragma unroll
      for (int v = 0; v < 8; ++v)
        acc[mi][ni][v] = inc;

  // Issue all four tile DMAs immediately; TENSORcnt completes in-order per wave.
  if (tid == 0) {
    v4i z4 = {0, 0, 0, 0};
    v8i z8 = {0, 0, 0, 0, 0, 0, 0, 0};
#pragma unroll
    for (int kc = 0; kc < K_DIM / KC; ++kc) {
      const float* aTile = A + (size_t)mBase * K_DIM + (size_t)kc * KC;
      const float* bTile = B + (size_t)kc * KC * N_DIM + nBase;
      v4u g0a = tdm_group0((uint32_t)(uintptr_t)&As[kc][0], aTile);
      v8i g1a = tdm_group1(/*td0*/KC, /*td1*/BM, /*tile0*/KC, /*tile1*/BM,
                           /*stride0*/K_DIM, /*interval 128dw*/6, /*pad 4dw*/3);
      v4u g0b = tdm_group0((uint32_t)(uintptr_t)&Bs[kc][0], bTile);
      v8i g1b = tdm_group1(/*td0*/BN, /*td1*/KC, /*tile0*/BN, /*tile1*/KC,
                           /*stride0*/N_DIM, /*interval 128dw*/6, /*pad 8dw*/7);
      __builtin_amdgcn_tensor_load_to_lds(g0a, g1a, z4, z4, z8, 0);
      __builtin_amdgcn_tensor_load_to_lds(g0b, g1b, z4, z4, z8, 0);
    }
  }

  // Chunk 0 ready when <=2 TDM ops remain outstanding (in-order completion).
  __builtin_amdgcn_s_wait_tensorcnt(2);
  __syncthreads();
  compute_chunk(&As[0][0], &Bs[0][0], waveM, waveN, l15, kh, acc);

  // Chunk 1 DMA overlapped with the compute above; now drain it.
  __builtin_amdgcn_s_wait_tensorcnt(0);
  __syncthreads();
  compute_chunk(&As[1][0], &Bs[1][0], waveM, waveN, l15, kh, acc);

  // C/D layout: VGPR v -> M = v (lanes 0-15) / v+8 (lanes 16-31), N = lane&15.
  // C is a 256 MB pure stream: store non-temporal so it doesn't wash L2.
#pragma unroll
  for (int mi = 0; mi < 2; ++mi) {
#pragma unroll
    for (int ni = 0; ni < 4; ++ni) {
      const int col = nBase + waveN * 64 + ni * 16 + l15;
#pragma unroll
      for (int v = 0; v < 8; ++v) {
        const int row = mBase + waveM * 32 + mi * 16 + v + 8 * kh;
        __builtin_nontemporal_store(acc[mi][ni][v], &C[(size_t)row * N_DIM + col]);
      }
    }
  }
}

extern "C" void kernel_launch(void* const* d_in, const int* in_sizes, int n_in,
                              void* d_out, int out_size, void* d_ws, size_t ws_size,
                              hipStream_t stream) {
  (void)in_sizes; (void)n_in; (void)out_size; (void)d_ws; (void)ws_size;
  const float* A = (const float*)d_in[0];
  const float* B = (const float*)d_in[1];
  const int*   w = (const int*)d_in[2];
  float*       C = (float*)d_out;
  dim3 grid(N_DIM / BN, M_DIM / BM);   // 64 x 64 workgroups
  PipelinedGemm_45664092291686_kernel<<<grid, dim3(256), 0, stream>>>(A, B, w, C);
}